// MyEmbedding_30855045055187
// MI455X (gfx1250) — compile-verified
//
#include <hip/hip_runtime.h>

// ---------------------------------------------------------------------------
// MyEmbedding gather for MI455X (gfx1250, wave32).
//
// out[b,l,:] = weight[id]                if id <  V
//            = figure_token_weight[id-V] if V <= id < V+NUM_FIG
//            = vis_emb[b, id-V-NUM_FIG]  otherwise
//
// Pure bandwidth problem: ~335 MB total traffic -> ~14.4 us at 23.3 TB/s.
// Grid (L, B): one workgroup (256 threads = 8 wave32) per output row; each
// thread moves exactly 5 x float4 (b128), fully unrolled. 2D grid removes the
// scalar integer-division sequence for b = row / L.
// ---------------------------------------------------------------------------

#define D_DIM   5120
#define V_DIM   32000
#define NUM_FIG 2
#define T_DIM   32

typedef float v4f __attribute__((ext_vector_type(4)));

__global__ __launch_bounds__(256, 1) void embed_gather_kernel(
    const int*   __restrict__ text_input,  // [B*L]
    const float* __restrict__ weight,      // [V, D]
    const float* __restrict__ fig,         // [NUM_FIG, D]
    const float* __restrict__ vis,         // [B, T, D]
    float*       __restrict__ out,         // [B*L, D]
    int L)
{
    const int b   = blockIdx.y;
    const int row = b * L + blockIdx.x;    // 0 .. B*L-1, no division
    const int id  = text_input[row];       // block-uniform -> scalar load

    // Block-uniform branch: select source row base.
    const float* src;
    if (id < V_DIM) {
        src = weight + (long long)id * D_DIM;
    } else if (id < V_DIM + NUM_FIG) {
        src = fig + (long long)(id - V_DIM) * D_DIM;
    } else {
        src = vis + ((long long)b * T_DIM + (id - V_DIM - NUM_FIG)) * (long long)D_DIM;
    }
    float* dst = out + (long long)row * D_DIM;

    const v4f* __restrict__ s4 = (const v4f*)src;
    v4f*       __restrict__ d4 = (v4f*)dst;

    const int t = threadIdx.x;

    // Speculative prefetch of this thread's first cacheline of the source row
    // (emits gfx1250 global_prefetch_b8).
    __builtin_prefetch(src + (size_t)t * 4, 0, 3);

    // D_DIM/4 = 1280 float4 per row; 1280 / 256 threads = 5 vectors/thread.
    // Fully unrolled so all 5 global_load_b128 are in flight simultaneously.
    v4f r0 = s4[t + 0 * 256];
    v4f r1 = s4[t + 1 * 256];
    v4f r2 = s4[t + 2 * 256];
    v4f r3 = s4[t + 3 * 256];
    v4f r4 = s4[t + 4 * 256];

    // Non-temporal stores: output is write-once; keep the 192 MB L2 for the
    // embedding table (token repetition gives real load-side reuse).
    __builtin_nontemporal_store(r0, &d4[t + 0 * 256]);
    __builtin_nontemporal_store(r1, &d4[t + 1 * 256]);
    __builtin_nontemporal_store(r2, &d4[t + 2 * 256]);
    __builtin_nontemporal_store(r3, &d4[t + 3 * 256]);
    __builtin_nontemporal_store(r4, &d4[t + 4 * 256]);
}

extern "C" void kernel_launch(void* const* d_in, const int* in_sizes, int n_in,
                              void* d_out, int out_size, void* d_ws, size_t ws_size,
                              hipStream_t stream) {
    // setup_inputs() order: text_input[B*L] i32, weight[V*D] f32,
    //                       figure_token_weight[NUM_FIG*D] f32, vis_emb[B*T*D] f32
    const int*   text   = (const int*)d_in[0];
    const float* weight = (const float*)d_in[1];
    const float* fig    = (const float*)d_in[2];
    const float* vis    = (const float*)d_in[3];
    float*       out    = (float*)d_out;

    const int BL = in_sizes[0];                          // B*L
    const int B  = in_sizes[3] / (T_DIM * D_DIM);        // from vis_emb size
    const int L  = BL / B;

    dim3 grid(L, B, 1);
    embed_gather_kernel<<<grid, 256, 0, stream>>>(text, weight, fig, vis, out, L);
}